// CausalSelfAttention2_24343874633880
// MI455X (gfx1250) — compile-verified
//
#include <hip/hip_runtime.h>
#include <hip/hip_bf16.h>

typedef float v2f __attribute__((ext_vector_type(2)));
typedef float v8f __attribute__((ext_vector_type(8)));

#define GROUPS 64            // B*H*ng = 2*4*8
#define GLEN   1025          // 1024 + mean row
#define HSZ    32
#define SCALE  0.17677669529663687f  // 1/sqrt(32)
#define PSTRIDE 68           // LDS row stride (even -> 8B aligned b64 reads)

__device__ __forceinline__ v8f wmma_f32(v2f a, v2f b, v8f c) {
    return __builtin_amdgcn_wmma_f32_16x16x4_f32(false, a, false, b, (short)0, c, false, false);
}

// ---------------------------------------------------------------------------
// Kernel 1: qkv = x @ w_attn.T, scattered into grouped layout (rows 0..1023).
// One wave computes one 16x16 output tile. M=16384, N=384, K=128.
// ---------------------------------------------------------------------------
__global__ __launch_bounds__(32) void qkv_gemm_kernel(
    const float* __restrict__ x, const float* __restrict__ w,
    float* __restrict__ qg, float* __restrict__ kg, float* __restrict__ vg)
{
    const int mt = blockIdx.x;            // 0..1023
    const int nt = blockIdx.y;            // 0..23
    const int lane = threadIdx.x;
    const int half = lane >> 4, l16 = lane & 15;

    const float* arow = x + (size_t)(mt * 16 + l16) * 128 + 2 * half;
    const float* brow = w + (size_t)(nt * 16 + l16) * 128 + 2 * half;

    v8f acc = {};
#pragma unroll
    for (int c = 0; c < 32; ++c) {
        float2 af = *(const float2*)(arow + 4 * c);
        float2 bf = *(const float2*)(brow + 4 * c);
        v2f a = { af.x, af.y };
        v2f b = { bf.x, bf.y };
        acc = wmma_f32(a, b, acc);
    }

    // scatter: column j = nt*16 + l16 -> which of q/k/v, head h, feature d
    const int j0 = nt * 16;
    const int which = j0 / 128;                    // 0=q,1=k,2=v (tile never straddles)
    float* dst = (which == 0) ? qg : ((which == 1) ? kg : vg);
    const int j = (j0 % 128) + l16;
    const int h = j / 32, d = j % 32;
    const int row0 = mt * 16;
    const int bidx = row0 / 8192;
#pragma unroll
    for (int r = 0; r < 8; ++r) {
        int t = (row0 % 8192) + r + 8 * half;      // token within batch
        int g = t / 1024, rig = t % 1024;          // group, row-in-group
        size_t off = ((size_t)((bidx * 4 + h) * 8 + g) * GLEN + rig) * HSZ + d;
        dst[off] = acc[r];
    }
}

// ---------------------------------------------------------------------------
// Kernel 2: fill row 1024 of each group with the mean of rows 0..1023.
// ---------------------------------------------------------------------------
__global__ __launch_bounds__(128) void mean_kernel(
    float* __restrict__ qg, float* __restrict__ kg, float* __restrict__ vg)
{
    const int grp = blockIdx.x;
    const int t = threadIdx.x;
    if (t >= 96) return;
    float* buf = (t < 32) ? qg : ((t < 64) ? kg : vg);
    const int d = t % 32;
    const float* base = buf + (size_t)grp * GLEN * HSZ + d;
    float s = 0.0f;
    for (int r = 0; r < 1024; ++r) s += base[r * HSZ];
    buf[(size_t)grp * GLEN * HSZ + 1024 * HSZ + d] = s * (1.0f / 1024.0f);
}

// ---------------------------------------------------------------------------
// Kernel 3: per-group causal flash attention, xg = v + softmax(QK^T/sqrt(32))V.
// One wave per 16-row Q tile; key blocks of 64 (4 column tiles per softmax
// pass) to amortize shuffles/rescales/barriers over 64 WMMAs per block.
// ---------------------------------------------------------------------------
__global__ __launch_bounds__(32) void attn_group_kernel(
    const float* __restrict__ qg, const float* __restrict__ kg,
    const float* __restrict__ vg, float* __restrict__ xg)
{
    const int qt = blockIdx.x;            // 0..64
    const int grp = blockIdx.y;           // 0..63
    const int lane = threadIdx.x;
    const int half = lane >> 4, l16 = lane & 15;

    const float* Q = qg + (size_t)grp * GLEN * HSZ;
    const float* K = kg + (size_t)grp * GLEN * HSZ;
    const float* V = vg + (size_t)grp * GLEN * HSZ;
    float*       X = xg + (size_t)grp * GLEN * HSZ;

    const int qbase = qt * 16;
    int qrow = qbase + l16; if (qrow > 1024) qrow = 1024;   // clamp (in-bounds reads)
    const float* qptr = Q + qrow * HSZ + 2 * half;

    v2f qf[8];
#pragma unroll
    for (int c = 0; c < 8; ++c) {
        float2 f = *(const float2*)(qptr + 4 * c);
        qf[c] = (v2f){ f.x, f.y };
    }

    v8f acc0 = {}, acc1 = {};             // O, feature halves 0..15 / 16..31
    float rowM[8], rowL[8];
#pragma unroll
    for (int r = 0; r < 8; ++r) { rowM[r] = -3.0e38f; rowL[r] = 0.0f; }

    __shared__ float lds[16 * PSTRIDE];   // 16x64 P staging

    for (int kbase = 0; kbase <= qbase + 15; kbase += 64) {
        // number of active 16-wide column tiles in this 64-key block (uniform)
        int jmax = ((qbase + 15 - kbase) >> 4) + 1;
        if (jmax > 4) jmax = 4;

        // ---- S = Q K^T for active column tiles (8 WMMA each) ----
        v8f s[4];
#pragma unroll
        for (int j = 0; j < 4; ++j) {
            s[j] = (v8f){};
            if (j < jmax) {                          // wave-uniform branch
                int krow = kbase + 16 * j + l16; if (krow > 1024) krow = 1024;
                const float* kptr = K + krow * HSZ + 2 * half;
#pragma unroll
                for (int c = 0; c < 8; ++c) {
                    float2 f = *(const float2*)(kptr + 4 * c);
                    v2f b = { f.x, f.y };
                    s[j] = wmma_f32(qf[c], b, s[j]);
                }
            }
        }

        // ---- one online-softmax pass over up to 64 columns ----
        float p[4][8];
#pragma unroll
        for (int r = 0; r < 8; ++r) {
            const int qi = qbase + r + 8 * half;
            float sv[4];
#pragma unroll
            for (int j = 0; j < 4; ++j) {
                const int kj = kbase + 16 * j + l16;
                const bool valid = (j < jmax) && (kj <= qi) && (kj <= 1024);
                sv[j] = valid ? s[j][r] * SCALE : -3.0e38f;
            }
            float rm = fmaxf(fmaxf(sv[0], sv[1]), fmaxf(sv[2], sv[3]));
            rm = fmaxf(rm, __shfl_xor(rm, 1, 32));
            rm = fmaxf(rm, __shfl_xor(rm, 2, 32));
            rm = fmaxf(rm, __shfl_xor(rm, 4, 32));
            rm = fmaxf(rm, __shfl_xor(rm, 8, 32));
            const float nm = fmaxf(rowM[r], rm);
            float rs = 0.0f;
#pragma unroll
            for (int j = 0; j < 4; ++j) {
                float pv = __expf(sv[j] - nm);       // masked: exp(-huge) -> +0
                p[j][r] = pv;
                rs += pv;
            }
            rs += __shfl_xor(rs, 1, 32);
            rs += __shfl_xor(rs, 2, 32);
            rs += __shfl_xor(rs, 4, 32);
            rs += __shfl_xor(rs, 8, 32);
            const float corr = __expf(rowM[r] - nm); // first block: exp(-huge)=0, acc=0
            rowM[r] = nm;
            rowL[r] = rowL[r] * corr + rs;
            acc0[r] = acc0[r] * corr;
            acc1[r] = acc1[r] * corr;
        }

        // ---- stage P (C/D layout) -> row-major LDS ----
        __syncthreads();
#pragma unroll
        for (int j = 0; j < 4; ++j)
#pragma unroll
            for (int r = 0; r < 8; ++r)
                lds[(r + 8 * half) * PSTRIDE + 16 * j + l16] = p[j][r];
        __syncthreads();

        // ---- O += P @ V over active keys (2 WMMA per K-chunk of 4) ----
        const int cmax = 4 * jmax;                   // uniform
#pragma unroll
        for (int c = 0; c < 16; ++c) {
            if (c < cmax) {                          // wave-uniform branch
                float2 pf = *(const float2*)&lds[l16 * PSTRIDE + 4 * c + 2 * half];
                v2f a = { pf.x, pf.y };
                int k0 = kbase + 4 * c + 2 * half;
                int kr0 = (k0     > 1024) ? 1024 : k0;   // masked cols have p==0
                int kr1 = (k0 + 1 > 1024) ? 1024 : k0 + 1;
                v2f b0 = { V[kr0 * HSZ + l16],      V[kr1 * HSZ + l16] };
                v2f b1 = { V[kr0 * HSZ + 16 + l16], V[kr1 * HSZ + 16 + l16] };
                acc0 = wmma_f32(a, b0, acc0);
                acc1 = wmma_f32(a, b1, acc1);
            }
        }
    }

    // xg = v + O / l
#pragma unroll
    for (int r = 0; r < 8; ++r) {
        const int row = qbase + r + 8 * half;
        if (row <= 1024) {
            const float invl = 1.0f / rowL[r];
            X[row * HSZ + l16]      = acc0[r] * invl + V[row * HSZ + l16];
            X[row * HSZ + 16 + l16] = acc1[r] * invl + V[row * HSZ + 16 + l16];
        }
    }
}

// ---------------------------------------------------------------------------
// Kernel 4: tiny causal attention over the 8 group-mean tokens per (b,h).
// ---------------------------------------------------------------------------
__global__ __launch_bounds__(256) void meanatt_kernel(
    const float* __restrict__ qg, const float* __restrict__ kg,
    const float* __restrict__ xg, float* __restrict__ y)
{
    const int bh = blockIdx.x;            // 0..7
    __shared__ float qm[8][32], km[8][32], vm[8][32];
    const int t = threadIdx.x;
    const int g = t / 32, d = t % 32;
    const size_t rowoff = ((size_t)(bh * 8 + g) * GLEN + 1024) * HSZ + d;
    qm[g][d] = qg[rowoff];
    km[g][d] = kg[rowoff];
    vm[g][d] = xg[rowoff];
    __syncthreads();

    float sarr[8];
    float mx = -3.0e38f;
    for (int gp = 0; gp <= g; ++gp) {
        float sv = 0.0f;
        for (int c = 0; c < 32; ++c) sv += qm[g][c] * km[gp][c];
        sv *= SCALE;
        sarr[gp] = sv;
        mx = fmaxf(mx, sv);
    }
    float denom = 0.0f;
    for (int gp = 0; gp <= g; ++gp) { sarr[gp] = __expf(sarr[gp] - mx); denom += sarr[gp]; }
    float o = 0.0f;
    for (int gp = 0; gp <= g; ++gp) o += sarr[gp] * vm[gp][d];
    y[(bh * 8 + g) * 32 + d] = o / denom;
}

// ---------------------------------------------------------------------------
// Kernel 5: reassemble z (B, 8200, 128) from xg and y.
// ---------------------------------------------------------------------------
__global__ __launch_bounds__(256) void assemble_kernel(
    const float* __restrict__ xg, const float* __restrict__ y, float* __restrict__ z)
{
    const int idx = blockIdx.x * 256 + threadIdx.x;     // < 2*8200*128
    if (idx >= 2 * 8200 * 128) return;
    const int c = idx % 128;
    const int t = (idx / 128) % 8200;
    const int b = idx / (8200 * 128);
    const int h = c / 32, d = c % 32;
    const int g = t / GLEN, row = t % GLEN;
    const int bh = b * 4 + h;
    float v;
    if (g == 0)        v = xg[((size_t)(bh * 8 + 0) * GLEN + row) * HSZ + d];
    else if (row == 0) v = y[(bh * 8 + (g - 1)) * 32 + d];
    else               v = xg[((size_t)(bh * 8 + g) * GLEN + (row - 1)) * HSZ + d];
    z[idx] = v;
}

// ---------------------------------------------------------------------------
// Kernel 6: out = z @ w_proj.T   (16400 x 128) * (128 x 128)
// ---------------------------------------------------------------------------
__global__ __launch_bounds__(32) void proj_gemm_kernel(
    const float* __restrict__ z, const float* __restrict__ w, float* __restrict__ out)
{
    const int mt = blockIdx.x;            // 0..1024 (16400 = 16*1025)
    const int nt = blockIdx.y;            // 0..7
    const int lane = threadIdx.x;
    const int half = lane >> 4, l16 = lane & 15;

    const float* arow = z + (size_t)(mt * 16 + l16) * 128 + 2 * half;
    const float* brow = w + (size_t)(nt * 16 + l16) * 128 + 2 * half;

    v8f acc = {};
#pragma unroll
    for (int c = 0; c < 32; ++c) {
        float2 af = *(const float2*)(arow + 4 * c);
        float2 bf = *(const float2*)(brow + 4 * c);
        v2f a = { af.x, af.y };
        v2f b = { bf.x, bf.y };
        acc = wmma_f32(a, b, acc);
    }
#pragma unroll
    for (int r = 0; r < 8; ++r) {
        const int row = mt * 16 + r + 8 * half;
        out[(size_t)row * 128 + nt * 16 + l16] = acc[r];
    }
}

// ---------------------------------------------------------------------------
extern "C" void kernel_launch(void* const* d_in, const int* in_sizes, int n_in,
                              void* d_out, int out_size, void* d_ws, size_t ws_size,
                              hipStream_t stream) {
    const float* x      = (const float*)d_in[0];
    const float* w_attn = (const float*)d_in[1];
    const float* w_proj = (const float*)d_in[2];

    float* ws = (float*)d_ws;
    const size_t GBUF = (size_t)GROUPS * GLEN * HSZ;    // 2,099,200 floats
    float* qg = ws;
    float* kg = ws + GBUF;
    float* vg = ws + 2 * GBUF;
    float* xg = ws + 3 * GBUF;
    float* y  = ws + 4 * GBUF;                          // 2048 floats
    float* z  = qg;                                     // alias: qg dead after meanatt

    qkv_gemm_kernel<<<dim3(1024, 24), 32, 0, stream>>>(x, w_attn, qg, kg, vg);
    mean_kernel<<<GROUPS, 128, 0, stream>>>(qg, kg, vg);
    attn_group_kernel<<<dim3(65, GROUPS), 32, 0, stream>>>(qg, kg, vg, xg);
    meanatt_kernel<<<8, 256, 0, stream>>>(qg, kg, xg, y);
    assemble_kernel<<<8200, 256, 0, stream>>>(xg, y, z);
    proj_gemm_kernel<<<dim3(1025, 8), 32, 0, stream>>>(z, w_proj, (float*)d_out);
}